// HMultiheadAttention_73813307949486
// MI455X (gfx1250) — compile-verified
//
#include <hip/hip_runtime.h>
#include <cmath>

typedef __attribute__((ext_vector_type(2))) float v2f;
typedef __attribute__((ext_vector_type(8))) float v8f;

#define EMBED  512
#define HEADS  8
#define HD     64
#define LSEQ   1024
#define NBATCH 4
#define NROWS  4096
#define EPSF   1e-7f
#define BOUND  (1.0f - 1e-7f)
#define PROJ_MAX (1.0f - 1e-5f)

// workspace layout (float indices)
#define WS_ZN    0
#define WS_CH    (3 * EMBED)
#define WS_SH    (6 * EMBED)
#define WS_HEADS (9 * EMBED)
#define HEADBUF  (HEADS * NBATCH * LSEQ * HD)   // 2,097,152 floats per tensor

// fast HW approximations (v_rcp_f32 / v_sqrt_f32 / v_exp_f32 / v_log_f32)
#define FRCP(x)  __builtin_amdgcn_rcpf(x)
#define FSQRT(x) __builtin_amdgcn_sqrtf(x)

// CDNA5 async global->LDS copy (ASYNCcnt tracked, no VGPR staging)
#define ASYNC_B128(lds32, ga64)                                        \
  asm volatile("global_load_async_to_lds_b128 %0, %1, off"             \
               :: "v"(lds32), "v"(ga64) : "memory")
#define WAIT_ASYNC() asm volatile("s_wait_asynccnt 0x0" ::: "memory")

static __device__ __forceinline__ v8f v8zero() {
  v8f r;
#pragma unroll
  for (int i = 0; i < 8; ++i) r[i] = 0.0f;
  return r;
}

// fp32 WMMA: D(16x16) = A(16x4) * B(4x16) + C   -> v_wmma_f32_16x16x4_f32
static __device__ __forceinline__ v8f wmma4(v2f a, v2f b, v8f c) {
  return __builtin_amdgcn_wmma_f32_16x16x4_f32(false, a, false, b, (short)0, c,
                                               false, false);
}

static __device__ __forceinline__ float rsum16(float v) {
#pragma unroll
  for (int off = 1; off < 16; off <<= 1) v += __shfl_xor(v, off, 16);
  return v;
}
static __device__ __forceinline__ float rmax16(float v) {
#pragma unroll
  for (int off = 1; off < 16; off <<= 1) v = fmaxf(v, __shfl_xor(v, off, 16));
  return v;
}

// asinh(u) = sign(u) * log(|u| + sqrt(u^2+1)) with fast sqrt/log
static __device__ __forceinline__ float fast_asinh(float u) {
  float au = fabsf(u);
  float r = __logf(au + FSQRT(__builtin_fmaf(au, au, 1.0f)));
  return copysignf(r, u);
}
// sinh(v) = 0.5*(e^v - e^-v), e^-v via rcp
static __device__ __forceinline__ float fast_sinh(float v) {
  float e = __expf(v);
  return 0.5f * (e - FRCP(e));
}

// ---------------------------------------------------------------------------
// Kernel 0: per-column z norms and cosh/sinh(2r) for the three projections
// ---------------------------------------------------------------------------
__global__ __launch_bounds__(512)
void hmha_prep_kernel(const float* __restrict__ zq, const float* __restrict__ zk,
                      const float* __restrict__ zv, const float* __restrict__ bq,
                      const float* __restrict__ bk, const float* __restrict__ bv,
                      float* __restrict__ ws) {
  const int t = blockIdx.x;   // 0..2 tensor
  const int j = threadIdx.x;  // 0..511 output column
  const float* z = (t == 0) ? zq : ((t == 1) ? zk : zv);
  const float* r = (t == 0) ? bq : ((t == 1) ? bk : bv);
  float s = 0.0f;
  for (int e = 0; e < EMBED; ++e) {
    float v = z[(size_t)e * EMBED + j];
    s += v * v;
  }
  float tc = 2.0f * r[j];
  float ep = __expf(tc), em = FRCP(ep);
  ws[WS_ZN + t * EMBED + j] = FSQRT(s);
  ws[WS_CH + t * EMBED + j] = 0.5f * (ep + em);
  ws[WS_SH + t * EMBED + j] = 0.5f * (ep - em);
}

// ---------------------------------------------------------------------------
// Kernel 1: Poincare linear (HNN++) + beta-split heads + projection.
// WG: 16 rows x 512 cols; 8 waves, wave w owns head w (64 cols).
// grid = (256 row-blocks, 3 tensors)
// ---------------------------------------------------------------------------
__global__ __launch_bounds__(256)
void hmha_plin_kernel(const float* __restrict__ xq, const float* __restrict__ xk,
                      const float* __restrict__ xv, const float* __restrict__ zq,
                      const float* __restrict__ zk, const float* __restrict__ zv,
                      float* __restrict__ ws, float split_scale) {
  __shared__ float lds_x[16 * 516];   // padded stride avoids bank conflicts
  __shared__ float s_x2[16];
  __shared__ float s_wsq[16];

  const int tid = threadIdx.x;
  const int lane = tid & 31;
  const int wid = tid >> 5;
  const int it = blockIdx.y;
  const int row0 = blockIdx.x * 16;

  const float* x = (it == 0) ? xq : ((it == 1) ? xk : xv);
  const float* z = (it == 0) ? zq : ((it == 1) ? zk : zv);
  const float* wzn = ws + WS_ZN + it * EMBED;
  const float* wch = ws + WS_CH + it * EMBED;
  const float* wsh = ws + WS_SH + it * EMBED;
  float* Hb = ws + WS_HEADS + (size_t)it * HEADBUF;

  if (tid < 16) { s_x2[tid] = 0.0f; s_wsq[tid] = 0.0f; }
  __syncthreads();

  {  // stage x tile (16 x 512) + row |x|^2
    const int rr = tid >> 4;
    const int seg = tid & 15;
    const float* xr = x + (size_t)(row0 + rr) * EMBED + seg * 32;
    float* dst = &lds_x[rr * 516 + seg * 32];
    float a2 = 0.0f;
#pragma unroll
    for (int i = 0; i < 32; i += 4) {
      float4 v = *(const float4*)(xr + i);
      *(float4*)(dst + i) = v;
      a2 += v.x * v.x + v.y * v.y + v.z * v.z + v.w * v.w;
    }
    atomicAdd(&s_x2[rr], a2);
  }
  __syncthreads();

  const int m16 = lane & 15;
  const int hb = lane >> 4;     // half-wave
  const int koff = hb * 2;      // K offset within WMMA fragment
  const int col0 = wid * 64;

  v8f acc[4];
#pragma unroll
  for (int t = 0; t < 4; ++t) acc[t] = v8zero();

  for (int e0 = 0; e0 < EMBED; e0 += 4) {
    v2f a = *(const v2f*)&lds_x[m16 * 516 + e0 + koff];
    const float* zb = z + (size_t)(e0 + koff) * EMBED;
#pragma unroll
    for (int t = 0; t < 4; ++t) {
      const int c = col0 + t * 16 + m16;
      v2f b;
      b[0] = zb[c];
      b[1] = zb[EMBED + c];
      acc[t] = wmma4(a, b, acc[t]);
    }
  }

  // elementwise HNN++ chain; lane's rows are j + 8*hb
  float lamr[8];
#pragma unroll
  for (int j = 0; j < 8; ++j) lamr[j] = 2.0f * FRCP(1.0f - s_x2[j + hb * 8]);

  float psum[8];
#pragma unroll
  for (int j = 0; j < 8; ++j) psum[j] = 0.0f;

#pragma unroll
  for (int t = 0; t < 4; ++t) {
    const int c = col0 + t * 16 + m16;
    const float zn = wzn[c];
    const float ch = wch[c];
    const float sh = wsh[c];
    const float izn = FRCP(zn);
#pragma unroll
    for (int j = 0; j < 8; ++j) {
      float xz = acc[t][j] * izn;
      float u = lamr[j] * xz * ch - (lamr[j] - 1.0f) * sh;
      float vv = 2.0f * zn * fast_asinh(u);
      float w = fast_sinh(vv);
      acc[t][j] = w;
      psum[j] += w * w;
    }
  }
#pragma unroll
  for (int j = 0; j < 8; ++j) {
    float s = rsum16(psum[j]);
    if (m16 == 0) atomicAdd(&s_wsq[j + hb * 8], s);
  }
  __syncthreads();

  float invden[8];
#pragma unroll
  for (int j = 0; j < 8; ++j)
    invden[j] = split_scale * FRCP(1.0f + FSQRT(1.0f + s_wsq[j + hb * 8]));

  // beta-split + per-head projection (wave owns exactly one head)
  float p2[8];
#pragma unroll
  for (int j = 0; j < 8; ++j) p2[j] = 0.0f;
#pragma unroll
  for (int t = 0; t < 4; ++t)
#pragma unroll
    for (int j = 0; j < 8; ++j) {
      float y = acc[t][j] * invden[j];
      acc[t][j] = y;
      p2[j] += y * y;
    }
  float scr[8];
#pragma unroll
  for (int j = 0; j < 8; ++j) {
    float n = FSQRT(rsum16(p2[j]));
    scr[j] = (n > PROJ_MAX) ? (PROJ_MAX * FRCP(fmaxf(n, EPSF))) : 1.0f;
  }
#pragma unroll
  for (int j = 0; j < 8; ++j) {
    const int row = row0 + j + hb * 8;
    const int l = row >> 2;
    const int nb = row & 3;
    float* op = Hb + (((size_t)wid * NBATCH + nb) * LSEQ + l) * HD + m16;
#pragma unroll
    for (int t = 0; t < 4; ++t) op[t * 16] = acc[t][j] * scr[j];
  }
}

// ---------------------------------------------------------------------------
// Kernel 2: hyperbolic attention (flash-style, unnormalized weights since the
// softmax scale cancels in the gyromidpoint num/den ratio).
// WG: one (h,n) batch x 64 q-rows; 4 waves x 16 q-rows; k-blocks of 64.
// grid = (16 q-blocks, 32 batches), block = 128
// ---------------------------------------------------------------------------
__global__ __launch_bounds__(128)
void hmha_attn_kernel(const float* __restrict__ ws, float* __restrict__ out,
                      float cat_scale) {
  __shared__ float lds_q[64 * 68];
  __shared__ float lds_k[64 * 68];
  __shared__ float lds_v[64 * 68];      // becomes v*lam after staging
  __shared__ float lds_p[4 * 16 * 20];  // per-wave P transpose scratch
  __shared__ float s_qq[64];
  __shared__ float s_kk[64];
  __shared__ float s_vv[64];
  __shared__ float s_lm1[64];           // lam - 1

  const int tid = threadIdx.x;
  const int lane = tid & 31;
  const int wid = tid >> 5;
  const int m16 = lane & 15;
  const int hb = lane >> 4;
  const int koff = hb * 2;
  const int qblk = blockIdx.x;
  const int batch = blockIdx.y;   // h*NBATCH + n

  const float* Qh = ws + WS_HEADS + (size_t)batch * LSEQ * HD;
  const float* Kh = ws + WS_HEADS + HEADBUF + (size_t)batch * LSEQ * HD;
  const float* Vh = ws + WS_HEADS + 2 * (size_t)HEADBUF + (size_t)batch * LSEQ * HD;

  const int q0 = qblk * 64;
  const int srow = tid >> 1;      // staging row 0..63
  const int shalf = tid & 1;

  if (tid < 64) s_qq[tid] = 0.0f;
  __syncthreads();
  {  // stage Q block + |q|^2
    const float* qr = Qh + (size_t)(q0 + srow) * HD + shalf * 32;
    float* dst = &lds_q[srow * 68 + shalf * 32];
    float p = 0.0f;
#pragma unroll
    for (int i = 0; i < 32; i += 4) {
      float4 a = *(const float4*)(qr + i);
      *(float4*)(dst + i) = a;
      p += a.x * a.x + a.y * a.y + a.z * a.z + a.w * a.w;
    }
    atomicAdd(&s_qq[srow], p);
  }
  __syncthreads();

  const int qrow = wid * 16;
  float qqr[8];
#pragma unroll
  for (int j = 0; j < 8; ++j) qqr[j] = s_qq[qrow + j + hb * 8];

  float M[8], den[8];
#pragma unroll
  for (int j = 0; j < 8; ++j) { M[j] = -1e30f; den[j] = 0.0f; }
  v8f O[4];
#pragma unroll
  for (int t = 0; t < 4; ++t) O[t] = v8zero();

  for (int kb = 0; kb < LSEQ; kb += 64) {
    __syncthreads();  // previous iteration done with lds_k / lds_v / s_* arrays
    if (tid < 64) { s_kk[tid] = 0.0f; s_vv[tid] = 0.0f; }
    {  // async-stage K,V block straight into LDS (ASYNCcnt path)
      const float* kr = Kh + (size_t)(kb + srow) * HD + shalf * 32;
      const float* vr = Vh + (size_t)(kb + srow) * HD + shalf * 32;
      float* dk = &lds_k[srow * 68 + shalf * 32];
      float* dv = &lds_v[srow * 68 + shalf * 32];
#pragma unroll
      for (int i = 0; i < 32; i += 4) {
        ASYNC_B128((unsigned int)(uintptr_t)(dk + i),
                   (unsigned long long)(uintptr_t)(kr + i));
        ASYNC_B128((unsigned int)(uintptr_t)(dv + i),
                   (unsigned long long)(uintptr_t)(vr + i));
      }
      if (kb + 64 < LSEQ) {
        __builtin_prefetch(kr + 64 * HD, 0, 1);   // global_prefetch_b8
        __builtin_prefetch(vr + 64 * HD, 0, 1);
      }
      WAIT_ASYNC();
    }
    __syncthreads();
    {  // row norms |k|^2, |v|^2 from LDS
      const float* dk = &lds_k[srow * 68 + shalf * 32];
      const float* dv = &lds_v[srow * 68 + shalf * 32];
      float pk = 0.0f, pv = 0.0f;
#pragma unroll
      for (int i = 0; i < 32; i += 4) {
        float4 a = *(const float4*)(dk + i);
        pk += a.x * a.x + a.y * a.y + a.z * a.z + a.w * a.w;
        float4 b = *(const float4*)(dv + i);
        pv += b.x * b.x + b.y * b.y + b.z * b.z + b.w * b.w;
      }
      atomicAdd(&s_kk[srow], pk);
      atomicAdd(&s_vv[srow], pv);
    }
    __syncthreads();
    {  // v <- v*lam in place; store lam-1
      const float lam = 2.0f * FRCP(1.0f - s_vv[srow]);
      float* dv = &lds_v[srow * 68 + shalf * 32];
#pragma unroll
      for (int i = 0; i < 32; ++i) dv[i] *= lam;
      if (shalf == 0) s_lm1[srow] = lam - 1.0f;
    }
    __syncthreads();

    // S = Q . K^T  (16q x 64k), K-dim = 64
    v8f acc[4];
#pragma unroll
    for (int t = 0; t < 4; ++t) acc[t] = v8zero();
    for (int e0 = 0; e0 < HD; e0 += 4) {
      v2f a = *(const v2f*)&lds_q[(qrow + m16) * 68 + e0 + koff];
#pragma unroll
      for (int t = 0; t < 4; ++t) {
        v2f b = *(const v2f*)&lds_k[(t * 16 + m16) * 68 + e0 + koff];
        acc[t] = wmma4(a, b, acc[t]);
      }
    }

    float kkv[4], lm1v[4];
#pragma unroll
    for (int t = 0; t < 4; ++t) {
      kkv[t] = s_kk[t * 16 + m16];
      lm1v[t] = s_lm1[t * 16 + m16];
    }

    // hyperbolic distance -> sim; online row max
    float rmx[8];
#pragma unroll
    for (int j = 0; j < 8; ++j) rmx[j] = -1e30f;
#pragma unroll
    for (int t = 0; t < 4; ++t) {
#pragma unroll
      for (int j = 0; j < 8; ++j) {
        float a_ = -acc[t][j];
        float qq = qqr[j];
        float A1 = 1.0f + 2.0f * a_ + kkv[t];
        float B1 = 1.0f - qq;
        float num2 = A1 * A1 * qq + 2.0f * A1 * B1 * a_ + B1 * B1 * kkv[t];
        float dd = 1.0f + 2.0f * a_ + qq * kkv[t];
        float nr = FSQRT(fmaxf(num2 * FRCP(dd * dd), 0.0f) + EPSF);
        float cl = fminf(nr, BOUND);
        float sim = __logf((1.0f - cl) * FRCP(1.0f + cl));  // -2*atanh(cl)
        acc[t][j] = sim;
        rmx[j] = fmaxf(rmx[j], sim);
      }
    }
    float fac[8];
#pragma unroll
    for (int j = 0; j < 8; ++j) {
      float nm = fmaxf(M[j], rmax16(rmx[j]));
      fac[j] = __expf(M[j] - nm);
      M[j] = nm;
      den[j] *= fac[j];
    }
#pragma unroll
    for (int t = 0; t < 4; ++t)
#pragma unroll
      for (int j = 0; j < 8; ++j) O[t][j] *= fac[j];
#pragma unroll
    for (int t = 0; t < 4; ++t)
#pragma unroll
      for (int j = 0; j < 8; ++j) {
        float w = __expf(acc[t][j] - M[j]);
        acc[t][j] = w;
        den[j] += w * lm1v[t];
      }

    // O += P @ (v*lam); transpose P (C-layout -> A-layout) via LDS
#pragma unroll
    for (int ct = 0; ct < 4; ++ct) {
#pragma unroll
      for (int j = 0; j < 8; ++j)
        lds_p[wid * 320 + (j + 8 * hb) * 20 + m16] = acc[ct][j];
#pragma unroll
      for (int e0 = 0; e0 < 16; e0 += 4) {
        v2f a = *(const v2f*)&lds_p[wid * 320 + m16 * 20 + e0 + koff];
#pragma unroll
        for (int dt = 0; dt < 4; ++dt) {
          const int kr_ = ct * 16 + e0 + koff;
          v2f b;
          b[0] = lds_v[kr_ * 68 + dt * 16 + m16];
          b[1] = lds_v[(kr_ + 1) * 68 + dt * 16 + m16];
          O[dt] = wmma4(a, b, O[dt]);
        }
      }
    }
  }

  // finalize: midpoint = num/den, Mobius half, beta-cat, scatter to (L,N,E)
  float idt[8];
#pragma unroll
  for (int j = 0; j < 8; ++j) idt[j] = FRCP(rsum16(den[j]));
  float p2[8];
#pragma unroll
  for (int j = 0; j < 8; ++j) p2[j] = 0.0f;
#pragma unroll
  for (int t = 0; t < 4; ++t)
#pragma unroll
    for (int j = 0; j < 8; ++j) {
      float mm = O[t][j] * idt[j];
      O[t][j] = mm;
      p2[j] += mm * mm;
    }
  float scl[8];
#pragma unroll
  for (int j = 0; j < 8; ++j) {
    float n = fmaxf(FSQRT(rsum16(p2[j])), EPSF);
    float y = fminf(n, BOUND);
    float tf = y * FRCP(1.0f + FSQRT(fmaxf(1.0f - y * y, 0.0f)));  // tanh(atanh(y)/2)
    scl[j] = tf * FRCP(n) * cat_scale;
  }
  const int h = batch >> 2;
  const int nb = batch & 3;
#pragma unroll
  for (int j = 0; j < 8; ++j) {
    const int l = q0 + qrow + j + 8 * hb;
    float* op = out + ((size_t)l * NBATCH + nb) * EMBED + h * HD + m16;
#pragma unroll
    for (int t = 0; t < 4; ++t) op[t * 16] = O[t][j] * scl[j];
  }
}

// ---------------------------------------------------------------------------
extern "C" void kernel_launch(void* const* d_in, const int* in_sizes, int n_in,
                              void* d_out, int out_size, void* d_ws, size_t ws_size,
                              hipStream_t stream) {
  (void)in_sizes; (void)n_in; (void)out_size; (void)ws_size;
  const float* q  = (const float*)d_in[0];
  const float* k  = (const float*)d_in[1];
  const float* v  = (const float*)d_in[2];
  const float* zq = (const float*)d_in[3];
  const float* bq = (const float*)d_in[4];
  const float* zk = (const float*)d_in[5];
  const float* bk = (const float*)d_in[6];
  const float* zv = (const float*)d_in[7];
  const float* bv = (const float*)d_in[8];
  float* out = (float*)d_out;
  float* ws  = (float*)d_ws;

  // HNN++ beta-split / beta-concat scales
  const double b_hd = exp(lgamma(HD / 2.0) + lgamma(0.5) - lgamma(HD / 2.0 + 0.5));
  const double b_e  = exp(lgamma(EMBED / 2.0) + lgamma(0.5) - lgamma(EMBED / 2.0 + 0.5));
  const float split_scale = (float)(b_hd / b_e);
  const float cat_scale   = (float)(b_e / b_hd);

  hmha_prep_kernel<<<dim3(3), dim3(512), 0, stream>>>(zq, zk, zv, bq, bk, bv, ws);
  hmha_plin_kernel<<<dim3(NROWS / 16, 3), dim3(256), 0, stream>>>(
      q, k, v, zq, zk, zv, ws, split_scale);
  hmha_attn_kernel<<<dim3(LSEQ / 64, HEADS * NBATCH), dim3(128), 0, stream>>>(
      ws, out, cat_scale);
}